// SSDDetector_32229434589217
// MI455X (gfx1250) — compile-verified
//
#include <hip/hip_runtime.h>
#include <hip/hip_bf16.h>

// SSD detector post-processing for MI455X (gfx1250).
// Reference: valid = cumprod(score >= 0.5) along K  ==  k < first_fail_index.
// Kernel A (ssd_out_kernel): per (batch, segment) write [box*scale, score] for
//   k < k0, zeros after. Live tiles are staged into LDS with gfx1250 async DMA
//   (global_load_async_to_lds_b128 + s_wait_asynccnt); all output stores are
//   non-temporal b128 (write-once stream, never re-read on device).
// Kernel B (ssd_k0_kernel): per batch, find first k with !(score >= 0.5)
//   via an early-exit chunked scan + LDS atomicMin.

#define KDET   8192
#define NCOMP  5
#define REC_F  (KDET * NCOMP)      // 40960 floats per (b, c) slab
#define SEG    1024                // records per segment
#define SEG_F  (SEG * NCOMP)       // 5120 floats per segment tile
#define NSEG   (KDET / SEG)        // 8 segments per batch
#define NTHR   256

typedef __attribute__((ext_vector_type(4))) float v4f;   // native vector: legal
                                                         // for nontemporal builtins

__device__ __forceinline__ void async_copy_b128(unsigned lds_byte_off,
                                                const float* gaddr) {
    // Per-lane 16B DMA from the global address gaddr to the LDS byte offset lds_byte_off, tracked by ASYNCcnt.
    asm volatile("global_load_async_to_lds_b128 %0, %1, off"
                 :: "v"(lds_byte_off), "v"(gaddr) : "memory");
}

__device__ __forceinline__ void wait_async0() {
    asm volatile("s_wait_asynccnt 0" ::: "memory");
}

// ---------------- Kernel A: masked scale/rotate, streamed output ----------------
__global__ __launch_bounds__(NTHR) void ssd_out_kernel(const float* __restrict__ det,
                                                       const float* __restrict__ scale,
                                                       const unsigned* __restrict__ k0_arr,
                                                       float* __restrict__ out) {
    __shared__ float smem[SEG_F];    // 20 KB tile

    const unsigned blk = blockIdx.x;
    const unsigned b   = blk / NSEG;
    const unsigned seg = blk % NSEG;
    const unsigned t0  = seg * SEG;           // first record of this segment
    const unsigned tid = threadIdx.x;
    const unsigned k0  = k0_arr[b];

    // 16B-aligned bases: REC_F*4 and SEG_F*4 are multiples of 16.
    float* ob = out + (size_t)b * REC_F + (size_t)t0 * NCOMP;

    if (t0 >= k0) {
        // Entire segment is masked out: stream zeros, no input traffic.
        const v4f z = {0.f, 0.f, 0.f, 0.f};
#pragma unroll
        for (int i = 0; i < SEG_F / (4 * NTHR); ++i)   // 5 iterations
            __builtin_nontemporal_store(z, (v4f*)ob + tid + (unsigned)(NTHR * i));
        return;
    }

    const float* ib = det + ((size_t)b * 2 + 1) * (size_t)REC_F + (size_t)t0 * NCOMP;

    // Stage the whole 5120-float tile into LDS via async DMA (b128 per lane).
    const unsigned lds0 = (unsigned)(size_t)(void*)smem;  // low 32 bits = LDS offset
#pragma unroll
    for (int i = 0; i < SEG_F / (4 * NTHR); ++i) {        // 5 x b128 per thread
        unsigned idx = tid + (unsigned)(NTHR * i);        // float4 index in tile
        async_copy_b128(lds0 + 16u * idx, ib + 4u * idx);
    }
    wait_async0();
    __syncthreads();

    const float s0 = scale[0], s1 = scale[1], s2 = scale[2], s3 = scale[3];
    unsigned nvalid = k0 - t0;
    if (nvalid > SEG) nvalid = SEG;

    // Flat output index j in [0, 5120): out[j] for j = 5k+c is
    //   c<4 : in[5k + c + 1] * scale[c]   (box)
    //   c==4: in[5k + 0]                  (score)
    // masked by k < nvalid. Stores are coalesced 16B-aligned NT b128.
#pragma unroll
    for (int i = 0; i < SEG_F / (4 * NTHR); ++i) {
        const unsigned j = 4u * (tid + (unsigned)(NTHR * i));
        v4f o;
#pragma unroll
        for (int c = 0; c < 4; ++c) {
            const unsigned jj = j + (unsigned)c;
            const unsigned k  = jj / 5u;
            const unsigned c5 = jj - 5u * k;
            float v = 0.0f;
            if (k < nvalid) {
                if (c5 == 4u) {
                    v = smem[jj - 4u];                    // score passes through
                } else {
                    const float sc = (c5 == 0u) ? s0 : (c5 == 1u) ? s1
                                   : (c5 == 2u) ? s2 : s3;
                    v = smem[jj + 1u] * sc;               // box * scale
                }
            }
            o[c] = v;
        }
        __builtin_nontemporal_store(o, (v4f*)ob + tid + (unsigned)(NTHR * i));
    }
}

// ---------------- Kernel B: first failing index per batch ----------------
__global__ __launch_bounds__(NTHR) void ssd_k0_kernel(const float* __restrict__ det,
                                                      unsigned* __restrict__ k0_out) {
    const unsigned b = blockIdx.x;
    // class-1 slab: det[b, 1, :, :]
    const float* base = det + ((size_t)b * 2 + 1) * (size_t)REC_F;

    __shared__ unsigned s_min;
    if (threadIdx.x == 0) s_min = KDET;
    __syncthreads();

    const unsigned t = threadIdx.x;
    const unsigned CHUNK = NTHR * 8;   // 2048 records per pass

    for (unsigned c0 = 0; c0 < KDET; c0 += CHUNK) {
        unsigned local = KDET;
#pragma unroll
        for (int i = 0; i < 8; ++i) {
            unsigned k = c0 + t + (unsigned)(NTHR * i);
            float sc = base[(size_t)k * NCOMP];     // component 0 = score
            if (!(sc >= 0.5f)) local = min(local, k); // NaN-safe fail test
        }
        if (local < KDET) atomicMin(&s_min, local);
        __syncthreads();
        unsigned cur = s_min;        // uniform snapshot between barriers
        __syncthreads();
        if (cur < c0 + CHUNK) break; // global first-fail already inside scanned prefix
    }
    if (threadIdx.x == 0) k0_out[b] = s_min;
}

extern "C" void kernel_launch(void* const* d_in, const int* in_sizes, int n_in,
                              void* d_out, int out_size, void* d_ws, size_t ws_size,
                              hipStream_t stream) {
    const float* det   = (const float*)d_in[0];   // (B, 2, 8192, 5) f32
    const float* scale = (const float*)d_in[1];   // (4,) f32
    float* out         = (float*)d_out;           // (B, 8192, 5) f32
    unsigned* k0_ws    = (unsigned*)d_ws;         // B unsigneds of scratch

    const int B = in_sizes[0] / (2 * REC_F);      // 256 for the reference shapes

    ssd_k0_kernel<<<B, NTHR, 0, stream>>>(det, k0_ws);
    ssd_out_kernel<<<B * NSEG, NTHR, 0, stream>>>(det, scale, k0_ws, out);
}